// Model_36902359007680
// MI455X (gfx1250) — compile-verified
//
#include <hip/hip_runtime.h>

typedef __attribute__((ext_vector_type(2))) float v2f;
typedef __attribute__((ext_vector_type(8))) float v8f;

namespace {
constexpr int B_   = 64;
constexpr int L_   = 512;
constexpr int C_   = 321;
constexpr int N_   = 2000;
constexpr int PRED = 96;
constexpr int QTOT = 48;                  // 24+12+8+4 gathered rows per dict
constexpr int KTOT = 512 + 512 + QTOT + QTOT;   // 1120
constexpr int KB   = 160;                 // K-block staged in LDS (7 blocks)
constexpr int LDSS = KB + 2;              // padded LDS stride (bank-conflict-free)
// workspace layout (in floats)
constexpr size_t WCAT_OFF = 0;                       // 96*1120
constexpr size_t BIAS_OFF = (size_t)PRED * KTOT;     // 96
constexpr size_t X_OFF    = BIAS_OFF + 128;          // B*KTOT*C
}

__device__ __forceinline__ void decode_q(int qq, int& i, int& Q, int& q, int& g) {
    if (qq < 24)      { i = 0; Q = 24; q = qq;      g = 4;  }
    else if (qq < 36) { i = 1; Q = 12; q = qq - 24; g = 8;  }
    else if (qq < 44) { i = 2; Q = 8;  q = qq - 36; g = 12; }
    else              { i = 3; Q = 4;  q = qq - 44; g = 24; }
}

// Wcat[p,k] : k<512 -> (wfs[:, :96] @ wxs), k<1024 -> (wft[:, :96] @ wxt),
// else folded period weights (wf[:, 96:] @ rw_i), seasonal block then trend block.
__global__ void build_wcat(const float* __restrict__ wfs, const float* __restrict__ wft,
                           const float* __restrict__ wxs, const float* __restrict__ wxt,
                           const float* __restrict__ rws0, const float* __restrict__ rws1,
                           const float* __restrict__ rws2, const float* __restrict__ rws3,
                           const float* __restrict__ rwt0, const float* __restrict__ rwt1,
                           const float* __restrict__ rwt2, const float* __restrict__ rwt3,
                           float* __restrict__ wcat) {
    int idx = blockIdx.x * blockDim.x + threadIdx.x;
    if (idx >= PRED * KTOT) return;
    int p = idx / KTOT;
    int k = idx % KTOT;
    float acc = 0.f;
    if (k < 512) {
        for (int t = 0; t < PRED; ++t) acc += wfs[p * 192 + t] * wxs[t * 512 + k];
    } else if (k < 1024) {
        int l = k - 512;
        for (int t = 0; t < PRED; ++t) acc += wft[p * 192 + t] * wxt[t * 512 + l];
    } else {
        int qq = k - 1024;
        bool isT = qq >= QTOT;
        if (isT) qq -= QTOT;
        int i, Q, q, g; decode_q(qq, i, Q, q, g);
        const float* rw = isT ? (i == 0 ? rwt0 : i == 1 ? rwt1 : i == 2 ? rwt2 : rwt3)
                              : (i == 0 ? rws0 : i == 1 ? rws1 : i == 2 ? rws2 : rws3);
        const float* wf = isT ? wft : wfs;
        for (int t = 0; t < PRED; ++t) acc += wf[p * 192 + 96 + t] * rw[t * Q + q];
    }
    wcat[(size_t)p * KTOT + k] = acc;
}

__global__ void build_bias(const float* __restrict__ bxs, const float* __restrict__ bxt,
                           const float* __restrict__ bfs, const float* __restrict__ bft,
                           const float* __restrict__ wfs, const float* __restrict__ wft,
                           const float* __restrict__ rbs0, const float* __restrict__ rbs1,
                           const float* __restrict__ rbs2, const float* __restrict__ rbs3,
                           const float* __restrict__ rbt0, const float* __restrict__ rbt1,
                           const float* __restrict__ rbt2, const float* __restrict__ rbt3,
                           float* __restrict__ biasv) {
    int p = threadIdx.x;
    if (p >= PRED) return;
    float acc = bfs[p] + bft[p];
    for (int t = 0; t < PRED; ++t) {
        acc += wfs[p * 192 + t] * bxs[t];
        acc += wft[p * 192 + t] * bxt[t];
        acc += wfs[p * 192 + 96 + t] * (rbs0[t] + rbs1[t] + rbs2[t] + rbs3[t]);
        acc += wft[p * 192 + 96 + t] * (rbt0[t] + rbt1[t] + rbt2[t] + rbt3[t]);
    }
    biasv[p] = acc;
}

// Moving-average decomposition; writes xn_seasonal into X[:,0:512,:], xn_trend into X[:,512:1024,:]
__global__ void decompose(const float* __restrict__ x_enc, float* __restrict__ X) {
    int tid = blockIdx.x * blockDim.x + threadIdx.x;
    if (tid >= B_ * C_) return;
    int b = tid / C_;
    int c = tid % C_;
    const float* xr = x_enc + (size_t)b * L_ * C_ + c;   // element l at xr[l*C_]
    // window sum at l=0: 12 pad copies of x[0] + x[0..12]
    float sum = 13.f * xr[0];
    for (int j = 1; j <= 12; ++j) sum += xr[(size_t)j * C_];
    // trend at l = L-1: x[L-13 .. L-1] + 12 pad copies of x[L-1]
    float offx = xr[(size_t)(L_ - 1) * C_];
    float sl = 12.f * offx;
    for (int j = L_ - 13; j <= L_ - 1; ++j) sl += xr[(size_t)j * C_];
    const float trendLast = sl * (1.f / 25.f);

    float* Xs = X + (size_t)b * KTOT * C_ + c;
    float* Xt = Xs + (size_t)512 * C_;
    for (int l = 0; l < L_; ++l) {
        float tr  = sum * (1.f / 25.f);
        float xnt = tr - trendLast;
        float xv  = xr[(size_t)l * C_];
        Xt[(size_t)l * C_] = xnt;
        Xs[(size_t)l * C_] = (xv - offx) - xnt;
        int jp = (l + 13 < L_ - 1) ? l + 13 : L_ - 1;
        int jm = (l - 12 > 0) ? l - 12 : 0;
        sum += xr[(size_t)jp * C_] - xr[(size_t)jm * C_];
    }
}

// Gather pr rows (ret[i][b, q*g, c]) into X[:,1024:1072,:] (seasonal) and X[:,1072:1120,:] (trend)
__global__ void gather_rows(const float* __restrict__ rds, const float* __restrict__ rdt,
                            const int* __restrict__ index, float* __restrict__ X) {
    int tid = blockIdx.x * blockDim.x + threadIdx.x;
    if (tid >= B_ * QTOT * C_) return;
    int c  = tid % C_;
    int r  = tid / C_;
    int qq = r % QTOT;
    int b  = r / QTOT;
    int i, Q, q, g; decode_q(qq, i, Q, q, g);
    int n = index[b];
    size_t src = ((size_t)(i * N_ + n) * PRED + q * g) * C_ + c;
    X[((size_t)b * KTOT + 1024 + qq) * C_ + c]        = rds[src];
    X[((size_t)b * KTOT + 1024 + QTOT + qq) * C_ + c] = rdt[src];
}

// out[b,p,c] = sum_k Wcat[p,k]*X[b,k,c] + bias[p] + x_enc[b,L-1,c]
// block (32,6): 6 waves = 6 M-tiles; each wave computes a 16x32 tile (2 WMMA frags).
// B tile (KB x 32 cols) staged in LDS once per block, shared by all 6 waves.
// grid: (ceil(336/32)=11 N-supertiles, 64 batches)
__global__ __launch_bounds__(192)
void raft_gemm(const float* __restrict__ wcat, const float* __restrict__ biasv,
               const float* __restrict__ X, const float* __restrict__ x_enc,
               float* __restrict__ out) {
    __shared__ float tileB[32 * LDSS];    // column-major [c][k], padded stride

    const int lane  = threadIdx.x;
    const int mtile = threadIdx.y;
    const int tid   = mtile * 32 + lane;
    const int c0    = blockIdx.x * 32;
    const int b     = blockIdx.y;
    const int half  = lane >> 4;          // ISA 16x4 f32: lanes 16-31 hold K+2
    const int l16   = lane & 15;
    const int prow  = mtile * 16 + l16;   // A row (0..95)

    const float* Ap = wcat + (size_t)prow * KTOT + half * 2;
    const float* Xb = X + (size_t)b * KTOT * C_;
    const float* lds0 = tileB + (size_t)l16 * LDSS + half * 2;          // cols c0+l16
    const float* lds1 = tileB + (size_t)(l16 + 16) * LDSS + half * 2;   // cols c0+16+l16

    v8f acc0 = {0.f, 0.f, 0.f, 0.f, 0.f, 0.f, 0.f, 0.f};
    v8f acc1 = {0.f, 0.f, 0.f, 0.f, 0.f, 0.f, 0.f, 0.f};

    for (int kb = 0; kb < KTOT; kb += KB) {
        // cooperative stage of B tile: X[kb..kb+KB-1][c0..c0+31] -> tileB[c][k]
        for (int idx = tid; idx < KB * 32; idx += 192) {
            int k  = idx >> 5;
            int c  = idx & 31;
            int cc = c0 + c; if (cc > C_ - 1) cc = C_ - 1;   // clamp tail columns
            tileB[c * LDSS + k] = Xb[(size_t)(kb + k) * C_ + cc];
        }
        // prefetch next A K-block while waiting at the barrier
        if (kb + KB < KTOT) __builtin_prefetch(Ap + kb + KB, 0, 1);
        __syncthreads();

        const float* ap = Ap + kb;
#pragma unroll 4
        for (int k = 0; k < KB; k += 4) {
            v2f a  = *(const v2f*)(ap + k);        // global (L2-resident wcat)
            v2f b0 = *(const v2f*)(lds0 + k);      // ds_load_b64
            v2f b1 = *(const v2f*)(lds1 + k);      // ds_load_b64
            acc0 = __builtin_amdgcn_wmma_f32_16x16x4_f32(
                false, a, false, b0, (short)0, acc0, false, false);
            acc1 = __builtin_amdgcn_wmma_f32_16x16x4_f32(
                false, a, false, b1, (short)0, acc1, false, false);
        }
        __syncthreads();
    }

    const int c1 = c0 + l16;
    const int c2 = c0 + 16 + l16;
    if (c1 < C_) {
        const float off = x_enc[((size_t)b * L_ + (L_ - 1)) * C_ + c1];
#pragma unroll
        for (int r = 0; r < 8; ++r) {
            int p = mtile * 16 + half * 8 + r;     // D layout: vgpr r -> M = r (+8 hi half)
            out[((size_t)b * PRED + p) * C_ + c1] = acc0[r] + biasv[p] + off;
        }
    }
    if (c2 < C_) {
        const float off = x_enc[((size_t)b * L_ + (L_ - 1)) * C_ + c2];
#pragma unroll
        for (int r = 0; r < 8; ++r) {
            int p = mtile * 16 + half * 8 + r;
            out[((size_t)b * PRED + p) * C_ + c2] = acc1[r] + biasv[p] + off;
        }
    }
}

extern "C" void kernel_launch(void* const* d_in, const int* in_sizes, int n_in,
                              void* d_out, int out_size, void* d_ws, size_t ws_size,
                              hipStream_t stream) {
    const float* x_enc = (const float*)d_in[0];
    const int*   index = (const int*)d_in[1];
    const float* rds   = (const float*)d_in[2];
    const float* rdt   = (const float*)d_in[3];
    const float* wxs   = (const float*)d_in[4];
    const float* bxs   = (const float*)d_in[5];
    const float* wxt   = (const float*)d_in[6];
    const float* bxt   = (const float*)d_in[7];
    const float* wfs   = (const float*)d_in[8];
    const float* bfs   = (const float*)d_in[9];
    const float* wft   = (const float*)d_in[10];
    const float* bft   = (const float*)d_in[11];
    const float* rws0  = (const float*)d_in[12];
    const float* rbs0  = (const float*)d_in[13];
    const float* rwt0  = (const float*)d_in[14];
    const float* rbt0  = (const float*)d_in[15];
    const float* rws1  = (const float*)d_in[16];
    const float* rbs1  = (const float*)d_in[17];
    const float* rwt1  = (const float*)d_in[18];
    const float* rbt1  = (const float*)d_in[19];
    const float* rws2  = (const float*)d_in[20];
    const float* rbs2  = (const float*)d_in[21];
    const float* rwt2  = (const float*)d_in[22];
    const float* rbt2  = (const float*)d_in[23];
    const float* rws3  = (const float*)d_in[24];
    const float* rbs3  = (const float*)d_in[25];
    const float* rwt3  = (const float*)d_in[26];
    const float* rbt3  = (const float*)d_in[27];

    float* ws    = (float*)d_ws;
    float* wcat  = ws + WCAT_OFF;
    float* biasv = ws + BIAS_OFF;
    float* X     = ws + X_OFF;
    float* out   = (float*)d_out;

    build_wcat<<<(PRED * KTOT + 255) / 256, 256, 0, stream>>>(
        wfs, wft, wxs, wxt, rws0, rws1, rws2, rws3, rwt0, rwt1, rwt2, rwt3, wcat);
    build_bias<<<1, 128, 0, stream>>>(
        bxs, bxt, bfs, bft, wfs, wft,
        rbs0, rbs1, rbs2, rbs3, rbt0, rbt1, rbt2, rbt3, biasv);
    decompose<<<(B_ * C_ + 127) / 128, 128, 0, stream>>>(x_enc, X);
    gather_rows<<<(B_ * QTOT * C_ + 255) / 256, 256, 0, stream>>>(rds, rdt, index, X);

    dim3 grid((C_ + 31) / 32, B_);   // 11 x 64
    dim3 blk(32, 6);                 // 6 waves: all M-tiles of one 32-wide N-supertile
    raft_gemm<<<grid, blk, 0, stream>>>(wcat, biasv, X, x_enc, out);
}